// MoE_for_Pruning_26096221290517
// MI455X (gfx1250) — compile-verified
//
#include <hip/hip_runtime.h>
#include <math.h>

typedef __attribute__((ext_vector_type(2))) float v2f;
typedef __attribute__((ext_vector_type(8))) float v8f;

// ---- order-preserving float -> uint key (monotone: f1 < f2  <=>  key1 < key2)
__device__ __forceinline__ unsigned fkey(float f) {
  unsigned u = __float_as_uint(f);
  return (u & 0x80000000u) ? ~u : (u | 0x80000000u);
}
#define NEGINF_KEY 0x007FFFFFu  // fkey(-inf)

// ---------------------------------------------------------------- init
__global__ __launch_bounds__(256) void k_init(unsigned* mxkey, unsigned* diff,
                                              unsigned* hist, float* accum,
                                              int N_, int nhist) {
  int t = blockIdx.x * blockDim.x + threadIdx.x;
  if (t < N_) { mxkey[t] = NEGINF_KEY; diff[t] = 1u; }
  if (t < nhist) hist[t] = 0u;
  if (t < 8) accum[t] = 0.0f;
}

__global__ __launch_bounds__(256) void k_diff(const int* old_idx, unsigned* diff, int NOLD) {
  int t = blockIdx.x * blockDim.x + threadIdx.x;
  if (t < NOLD) diff[old_idx[t]] = 0u;
}

// compute K_l on device from input scalar l, init selection state
__global__ __launch_bounds__(128) void k_state(const int* l_ptr, unsigned* prefix,
                                               unsigned* kstate, int nsel) {
  __shared__ int Ks;
  if (threadIdx.x == 0) {
    int l = l_ptr[0];
    float K;
    if (l < 4) {
      float s = 1.0f / (1.0f + expf(-((float)l - 2.0f)));
      K = 100.0f + 900.0f * s;                 // K_SOURCE + (K_MAX-K_SOURCE)*s
    } else {
      float s = 1.0f / (1.0f + expf(-((float)l - 6.0f)));
      K = 50.0f + 950.0f * (1.0f - s);         // K_MIN + (K_MAX-K_MIN)*(1-s)
    }
    Ks = (int)K;
  }
  __syncthreads();
  if ((int)threadIdx.x < nsel) { prefix[threadIdx.x] = 0u; kstate[threadIdx.x] = (unsigned)Ks; }
}

// ---------------------------------------------------------------- tiny MLP with WMMA
// ctx = relu([hidden_q | rel_embed[q_rel]] @ W1 + b1) @ W2 + b2  (32x64)
// esc = ctx @ expert_emb^T (32x3); nscale = softplus(ctx @ w_n) (32)
__global__ __launch_bounds__(256) void k_ctx(const float* __restrict__ hidden_q,
                                             const int* __restrict__ q_rel,
                                             const float* __restrict__ rel_embed,
                                             const float* __restrict__ w1,
                                             const float* __restrict__ b1,
                                             const float* __restrict__ w2,
                                             const float* __restrict__ b2,
                                             const float* __restrict__ expert_emb,
                                             const float* __restrict__ w_n,
                                             float* qn_out, float* qnorm_out,
                                             float* esc_out, float* nscale_out) {
  __shared__ float A1[32][128];   // ctx_in ; also reused as CTX (32x64) later
  __shared__ float W[128][64];    // W1 then W2
  __shared__ float H[32][64];     // hidden layer
  const int tid = threadIdx.x;

  for (int idx = tid; idx < 32 * 128; idx += 256) {
    int r = idx >> 7, c = idx & 127;
    float v;
    if (c < 64) v = hidden_q[r * 64 + c];
    else        v = rel_embed[q_rel[r] * 64 + (c - 64)];
    A1[r][c] = v;
  }
  for (int idx = tid; idx < 128 * 64; idx += 256) W[idx >> 6][idx & 63] = w1[idx];
  __syncthreads();

  // export qn (= rel_embed[q_rel][b]) and its clamped norm
  for (int idx = tid; idx < 32 * 64; idx += 256)
    qn_out[idx] = A1[idx >> 6][64 + (idx & 63)];
  if (tid < 32) {
    float s = 0.f;
    for (int d = 0; d < 64; ++d) { float v = A1[tid][64 + d]; s += v * v; }
    qnorm_out[tid] = fmaxf(sqrtf(s), 1e-8f);
  }

  // --- GEMM1: (32x128)@(128x64), 8 waves -> 2x4 tiles of 16x16, K-step 4 (WMMA f32)
  const int wave = tid >> 5, lane = tid & 31;
  const int mt = wave & 1, nt = wave >> 1;
  const int mrow = (mt << 4) + (lane & 15);
  const int ncol = (nt << 4) + (lane & 15);
  const int khalf = (lane >> 4) << 1;           // 0 or 2
  const int base_m = (mt << 4) + ((lane >> 4) << 3);

  v8f acc = {};
  for (int k0 = 0; k0 < 128; k0 += 4) {
    v2f a, b;
    a.x = A1[mrow][k0 + khalf + 0];
    a.y = A1[mrow][k0 + khalf + 1];
    b.x = W[k0 + khalf + 0][ncol];
    b.y = W[k0 + khalf + 1][ncol];
    acc = __builtin_amdgcn_wmma_f32_16x16x4_f32(false, a, false, b, (short)0, acc,
                                                false, false);
  }
  {
    float bias = b1[ncol];
    for (int r = 0; r < 8; ++r) H[base_m + r][ncol] = fmaxf(acc[r] + bias, 0.0f);
  }
  __syncthreads();

  // --- GEMM2: (32x64)@(64x64)
  for (int idx = tid; idx < 64 * 64; idx += 256) W[idx >> 6][idx & 63] = w2[idx];
  __syncthreads();

  acc = (v8f){};
  for (int k0 = 0; k0 < 64; k0 += 4) {
    v2f a, b;
    a.x = H[mrow][k0 + khalf + 0];
    a.y = H[mrow][k0 + khalf + 1];
    b.x = W[k0 + khalf + 0][ncol];
    b.y = W[k0 + khalf + 1][ncol];
    acc = __builtin_amdgcn_wmma_f32_16x16x4_f32(false, a, false, b, (short)0, acc,
                                                false, false);
  }
  float (*CTX)[64] = (float(*)[64])A1;          // reuse A1 space
  {
    float bias = b2[ncol];
    for (int r = 0; r < 8; ++r) CTX[base_m + r][ncol] = acc[r] + bias;
  }
  __syncthreads();

  if (tid < 96) {
    int bb = tid / 3, e = tid % 3;
    float s = 0.f;
    for (int d = 0; d < 64; ++d) s += CTX[bb][d] * expert_emb[e * 64 + d];
    esc_out[bb * 3 + e] = s;
  } else if (tid < 128) {
    int bb = tid - 96;
    float s = 0.f;
    for (int d = 0; d < 64; ++d) s += CTX[bb][d] * w_n[d];
    nscale_out[bb] = (s > 20.f) ? s : log1pf(expf(s));   // softplus
  }
}

// ---------------------------------------------------------------- per-node softmax3 reductions
__global__ __launch_bounds__(256) void k_weights(const int* __restrict__ nodes,
                                                 const float* __restrict__ noise,
                                                 const float* __restrict__ esc,
                                                 const float* __restrict__ nscale,
                                                 float* accum, int N_) {
  int i = blockIdx.x * blockDim.x + threadIdx.x;
  float c0 = 0.f, c1 = 0.f, c2 = 0.f, tt = 0.f, ss = 0.f;
  if (i < N_) {
    int b = nodes[2 * i];
    float ns = nscale[b];                         // LAM_NOISE = 1, TEMP = 1
    float e0 = esc[b * 3 + 0] + noise[3 * i + 0] * ns;
    float e1 = esc[b * 3 + 1] + noise[3 * i + 1] * ns;
    float e2 = esc[b * 3 + 2] + noise[3 * i + 2] * ns;
    float m = fmaxf(e0, fmaxf(e1, e2));
    float x0 = expf(e0 - m), x1 = expf(e1 - m), x2 = expf(e2 - m);
    float s = x0 + x1 + x2;
    c0 = x0 / s; c1 = x1 / s; c2 = x2 / s;
    tt = c0 + c1 + c2;
    ss = c0 * c0 + c1 * c1 + c2 * c2;
  }
  for (int off = 16; off; off >>= 1) {
    c0 += __shfl_down(c0, off, 32);
    c1 += __shfl_down(c1, off, 32);
    c2 += __shfl_down(c2, off, 32);
    tt += __shfl_down(tt, off, 32);
    ss += __shfl_down(ss, off, 32);
  }
  __shared__ float red[8][5];
  int lane = threadIdx.x & 31, wv = threadIdx.x >> 5;
  if (lane == 0) { red[wv][0]=c0; red[wv][1]=c1; red[wv][2]=c2; red[wv][3]=tt; red[wv][4]=ss; }
  __syncthreads();
  if (threadIdx.x == 0) {
    float a0=0,a1=0,a2=0,a3=0,a4=0;
    for (int w = 0; w < 8; ++w) { a0+=red[w][0]; a1+=red[w][1]; a2+=red[w][2]; a3+=red[w][3]; a4+=red[w][4]; }
    atomicAdd(&accum[0], a0); atomicAdd(&accum[1], a1); atomicAdd(&accum[2], a2);
    atomicAdd(&accum[3], a3); atomicAdd(&accum[4], a4);
  }
}

// ---------------------------------------------------------------- cosine(hidden[i], qn[b])
__global__ __launch_bounds__(256) void k_cos(const float* __restrict__ hidden,
                                             const int* __restrict__ nodes,
                                             const float* __restrict__ qn,
                                             const float* __restrict__ qnorm,
                                             unsigned* coskey, int N_) {
  long long t = (long long)blockIdx.x * blockDim.x + threadIdx.x;
  if (t >= (long long)N_ * 16) return;             // 16 float4 segments per node (D=64)
  int node = (int)(t >> 4);
  int seg = (int)(t & 15);
  float4 hv = ((const float4*)hidden)[(long long)node * 16 + seg];
  int b = nodes[2 * node];
  float4 qv = ((const float4*)qn)[b * 16 + seg];
  float dot = hv.x*qv.x + hv.y*qv.y + hv.z*qv.z + hv.w*qv.w;
  float nn  = hv.x*hv.x + hv.y*hv.y + hv.z*hv.z + hv.w*hv.w;
  for (int off = 1; off < 16; off <<= 1) {
    dot += __shfl_xor(dot, off, 32);
    nn  += __shfl_xor(nn,  off, 32);
  }
  if (seg == 0) {
    float den = fmaxf(sqrtf(nn), 1e-8f) * qnorm[b];
    coskey[node] = fkey(dot / den);
  }
}

// ---------------------------------------------------------------- segment max over edges
__global__ __launch_bounds__(256) void k_segmax(const int* __restrict__ edges,
                                                const float* __restrict__ alpha_temp,
                                                unsigned* mxkey, int NE) {
  int e = blockIdx.x * blockDim.x + threadIdx.x;
  if (e >= NE) return;
  int obj = edges[e * 6 + 5];
  atomicMax(&mxkey[obj], fkey(alpha_temp[e]));
}

// ---------------------------------------------------------------- radix-select histogram pass
__global__ __launch_bounds__(256) void k_hist(const int* __restrict__ nodes,
                                              const float* __restrict__ scores,
                                              const unsigned* __restrict__ coskey,
                                              const unsigned* __restrict__ mxkey,
                                              const unsigned* __restrict__ diff,
                                              const unsigned* __restrict__ prefix,
                                              unsigned* hist, int pass, int N_, int B_) {
  int i = blockIdx.x * blockDim.x + threadIdx.x;
  if (i >= N_ || !diff[i]) return;
  int b = nodes[2 * i];
  unsigned keys[3] = { fkey(scores[i]), coskey[i], mxkey[i] };
  int sels[3] = { b, B_ + b, 2 * B_ };
  int sh = 24 - 8 * pass;
  #pragma unroll
  for (int j = 0; j < 3; ++j) {
    unsigned key = keys[j];
    bool ok = (pass == 0) || ((key >> (32 - 8 * pass)) == prefix[sels[j]]);
    if (ok) atomicAdd(&hist[sels[j] * 256 + ((key >> sh) & 255u)], 1u);
  }
}

__global__ __launch_bounds__(128) void k_scan(unsigned* hist, unsigned* prefix,
                                              unsigned* kstate, int nsel) {
  int sel = threadIdx.x;
  if (sel >= nsel) return;
  unsigned k = kstate[sel];
  unsigned chosen = 0;
  for (int bin = 255; bin >= 0; --bin) {
    unsigned c = hist[sel * 256 + bin];
    if (c >= k || bin == 0) { chosen = (unsigned)bin; break; }
    k -= c;
  }
  prefix[sel] = (prefix[sel] << 8) | chosen;
  kstate[sel] = k;
  for (int bin = 0; bin < 256; ++bin) hist[sel * 256 + bin] = 0u;  // ready for next pass
}

// ---------------------------------------------------------------- wm + scores_c + L_imp
__global__ __launch_bounds__(256) void k_wm(const int* __restrict__ nodes,
                                            const float* __restrict__ scores,
                                            const unsigned* __restrict__ coskey,
                                            const unsigned* __restrict__ mxkey,
                                            const unsigned* __restrict__ diff,
                                            const unsigned* __restrict__ prefix,
                                            const float* __restrict__ accum,
                                            float* wm, float* d_out,
                                            int N_, int D_, int B_) {
  int i = blockIdx.x * blockDim.x + threadIdx.x;
  if (i >= N_) return;
  float total = accum[3];
  float w0 = accum[0] / total, w1 = accum[1] / total, w2 = accum[2] / total;
  float m0 = 1.f, m1 = 1.f, m2 = 1.f;
  if (diff[i]) {
    int b = nodes[2 * i];
    m0 = (fkey(scores[i]) >= prefix[b])       ? 1.f : 0.f;
    m1 = (coskey[i]      >= prefix[B_ + b])   ? 1.f : 0.f;
    m2 = (mxkey[i]       >= prefix[2 * B_])   ? 1.f : 0.f;
  }
  float w = w0 * m0 + w1 * m1 + w2 * m2;
  wm[i] = w;
  d_out[(long long)N_ * D_ + i] = scores[i] * w;                 // scores_c
  if (i == 0) {
    float n = 3.0f * (float)N_;
    float mean = total / n;
    float var = (accum[4] - total * total / n) / (n - 1.0f);
    float cv = sqrtf(var) / (mean + 1e-5f);
    d_out[(long long)2 * N_ * D_ + N_] = cv * cv;                // L_imp
  }
}

// ---------------------------------------------------------------- hidden_c / h0_c streaming
__global__ __launch_bounds__(256) void k_scale(const float* __restrict__ hidden,
                                               const float* __restrict__ h0,
                                               const float* __restrict__ wm,
                                               float* d_out, int N_, int D_) {
  long long t = (long long)blockIdx.x * blockDim.x + threadIdx.x;
  int segs = D_ >> 2;
  if (t >= (long long)N_ * segs) return;
  int node = (int)(t / segs);
  float w = wm[node];
  float4 hv = ((const float4*)hidden)[t];
  float4 gv = ((const float4*)h0)[t];
  ((float4*)d_out)[t] = make_float4(hv.x * w, hv.y * w, hv.z * w, hv.w * w);
  float* h0c = d_out + (long long)N_ * D_ + N_;
  ((float4*)h0c)[t] = make_float4(gv.x * w, gv.y * w, gv.z * w, gv.w * w);
}

// ================================================================ host
extern "C" void kernel_launch(void* const* d_in, const int* in_sizes, int n_in,
                              void* d_out, int out_size, void* d_ws, size_t ws_size,
                              hipStream_t stream) {
  const float* hidden     = (const float*)d_in[0];
  const int*   nodes      = (const int*)d_in[1];
  const float* scores     = (const float*)d_in[2];
  const float* h0         = (const float*)d_in[3];
  const float* hidden_q   = (const float*)d_in[5];
  const int*   q_rel      = (const int*)d_in[6];
  const int*   edges      = (const int*)d_in[7];
  const int*   old_idx    = (const int*)d_in[8];
  const float* alpha_temp = (const float*)d_in[12];
  const int*   l_ptr      = (const int*)d_in[13];
  const float* noise      = (const float*)d_in[14];
  const float* mlp_w1     = (const float*)d_in[15];
  const float* mlp_b1     = (const float*)d_in[16];
  const float* mlp_w2     = (const float*)d_in[17];
  const float* mlp_b2     = (const float*)d_in[18];
  const float* rel_embed  = (const float*)d_in[19];
  const float* expert_emb = (const float*)d_in[20];
  const float* w_n        = (const float*)d_in[21];

  const int N  = in_sizes[2];
  const int D  = in_sizes[0] / N;        // 64
  const int B  = in_sizes[5] / D;        // 32
  const int NE = in_sizes[12];           // NEDGE
  const int NOLD = in_sizes[8];
  const int nsel = 2 * B + 1;            // 65
  const int nhist = nsel * 256;

  // workspace carve-up (all 4-byte words)
  unsigned* coskey = (unsigned*)d_ws;
  unsigned* mxkey  = coskey + N;
  float*    wm     = (float*)(mxkey + N);
  unsigned* diff   = (unsigned*)(wm + N);
  float*    qn     = (float*)(diff + N);
  float*    qnorm  = qn + B * D;
  float*    esc    = qnorm + B;
  float*    nscale = esc + 3 * B;
  float*    accum  = nscale + B;         // 8 slots
  unsigned* hist   = (unsigned*)(accum + 8);
  unsigned* prefix = hist + nhist;
  unsigned* kstate = prefix + nsel;

  float* out = (float*)d_out;
  const int T = 256;
  int gN = (N + T - 1) / T;

  k_init<<<gN, T, 0, stream>>>(mxkey, diff, hist, accum, N, nhist);
  k_diff<<<(NOLD + T - 1) / T, T, 0, stream>>>(old_idx, diff, NOLD);
  k_state<<<1, 128, 0, stream>>>(l_ptr, prefix, kstate, nsel);

  k_ctx<<<1, 256, 0, stream>>>(hidden_q, q_rel, rel_embed, mlp_w1, mlp_b1,
                               mlp_w2, mlp_b2, expert_emb, w_n,
                               qn, qnorm, esc, nscale);

  k_weights<<<gN, T, 0, stream>>>(nodes, noise, esc, nscale, accum, N);

  long long cosThreads = (long long)N * 16;
  k_cos<<<(int)((cosThreads + T - 1) / T), T, 0, stream>>>(hidden, nodes, qn, qnorm,
                                                           coskey, N);

  k_segmax<<<(NE + T - 1) / T, T, 0, stream>>>(edges, alpha_temp, mxkey, NE);

  for (int pass = 0; pass < 4; ++pass) {
    k_hist<<<gN, T, 0, stream>>>(nodes, scores, coskey, mxkey, diff, prefix,
                                 hist, pass, N, B);
    k_scan<<<1, 128, 0, stream>>>(hist, prefix, kstate, nsel);
  }

  k_wm<<<gN, T, 0, stream>>>(nodes, scores, coskey, mxkey, diff, prefix, accum,
                             wm, out, N, D, B);

  long long scaleThreads = (long long)N * (D / 4);
  k_scale<<<(int)((scaleThreads + T - 1) / T), T, 0, stream>>>(hidden, h0, wm,
                                                               out, N, D);
}